// SG16_3496103379566
// MI455X (gfx1250) — compile-verified
//
#include <hip/hip_runtime.h>
#include <hip/hip_bf16.h>

// SGConv x16 stack for MI455X (gfx1250, wave32).
//   1) Build CSR by dst once per launch (count -> scan -> atomic fill); each
//      CSR entry packs (src, norm) into one int2 so the aggregation loop does
//      a single coalesced 8B load per 32 edges per lane.
//   2) h0 = x @ w0 + b0 via V_WMMA_F32_16X16X4_F32 (fp32, K=128).
//   3) 16 fused layers: 32-node tile per 256-thread block; each wave gathers
//      4 nodes (lane == feature, coalesced 128B h-row loads from L2-resident
//      12.8MB h). Edges fetched cooperatively (1 global_load_b64 / 32 edges /
//      lane) and broadcast with v_readlane (SGPR result -> scalar-base gather
//      addressing, LDS pipe stays free). Then 4 waves do the 32x32 GEMM with
//      8 chained f32 WMMAs + bias + ReLU.
//   4) out = h @ w16 + b16 (8 waves, 32x64 tile, 8 chained WMMAs each).

typedef __attribute__((ext_vector_type(2))) float v2f;
typedef __attribute__((ext_vector_type(8))) float v8f;

#define DIN 128
#define HF  32
#define DOUT 64
#define NLAYERS 16

// ---------------- CSR build ----------------

__global__ void zero_cnt_kernel(int* cnt, int n) {
    int i = blockIdx.x * blockDim.x + threadIdx.x;
    if (i < n) cnt[i] = 0;
}

__global__ void count_kernel(const int* __restrict__ ei, int* cnt, int e) {
    int k = blockIdx.x * blockDim.x + threadIdx.x;
    if (k < e) atomicAdd(&cnt[ei[e + k]], 1);   // dst row of edge_index
}

__global__ void dinv_kernel(const int* __restrict__ cnt, float* dinv, int n) {
    int i = blockIdx.x * blockDim.x + threadIdx.x;
    if (i < n) dinv[i] = rsqrtf((float)(cnt[i] + 1));  // +1 self loop
}

// Single-block chunked exclusive scan of cnt -> row_ptr (and fill_ptr copy).
__global__ void scan_kernel(const int* __restrict__ cnt, int* row_ptr,
                            int* fill_ptr, int n) {
    __shared__ int sdata[256];
    __shared__ int s_running;
    int t = threadIdx.x;
    if (t == 0) s_running = 0;
    __syncthreads();
    for (int base = 0; base < n; base += 256) {
        int i = base + t;
        int v = (i < n) ? cnt[i] : 0;
        sdata[t] = v;
        __syncthreads();
        for (int off = 1; off < 256; off <<= 1) {
            int add = (t >= off) ? sdata[t - off] : 0;
            __syncthreads();
            sdata[t] += add;
            __syncthreads();
        }
        int incl = sdata[t];
        int excl = incl - v;
        int base_off = s_running;
        __syncthreads();
        if (t == 255) s_running = base_off + incl;  // chunk total
        __syncthreads();
        if (i < n) {
            int rp = base_off + excl;
            row_ptr[i]  = rp;
            fill_ptr[i] = rp;
        }
        __syncthreads();
    }
    if (t == 0) row_ptr[n] = s_running;
}

__global__ void fill_kernel(const int* __restrict__ ei,
                            const float* __restrict__ dinv,
                            int* fill_ptr, int2* edge, int e) {
    int k = blockIdx.x * blockDim.x + threadIdx.x;
    if (k >= e) return;
    int s = ei[k];
    int d = ei[e + k];
    int pos = atomicAdd(&fill_ptr[d], 1);
    edge[pos] = make_int2(s, __float_as_int(dinv[s] * dinv[d]));
}

// ---------------- WMMA fragment layout (f32 16x16x4) ----------------
// A (16x4): lane L holds M=L%16; K = 2*(L/16) + vgpr.
// B (4x16): lane L holds N=L%16; K = 2*(L/16) + vgpr (mirrored layout).
// C/D (16x16): vgpr r, lane L: M = r + 8*(L/16), N = L%16.

// ---------------- Input GEMM: h0 = x @ w0 + b0 ----------------
// block: 256 threads, 32 nodes; K = 128 -> 32 chained WMMAs per 16x16 tile.

__global__ void gemm_in_kernel(const float* __restrict__ x,
                               const float* __restrict__ w0,
                               const float* __restrict__ b0,
                               float* __restrict__ h0) {
    __shared__ float tile[32 * 129];   // 32x128, padded stride 129
    int t = threadIdx.x;
    int node0 = blockIdx.x * 32;
    for (int idx = t; idx < 32 * DIN; idx += 256) {
        int r = idx >> 7, c = idx & 127;
        tile[r * 129 + c] = x[(size_t)(node0 + r) * DIN + c];
    }
    __syncthreads();
    int wave = t >> 5, lane = t & 31;
    if (wave < 4) {
        int rowt = wave >> 1;              // 16-node row tile
        int colh = wave & 1;               // 16-col half
        int half = lane >> 4;
        int m    = lane & 15;
        int ncol = (colh << 4) + m;
        const float* at = tile + rowt * 16 * 129;
        v8f acc = {};
        for (int c = 0; c < DIN / 4; ++c) {
            int k0 = 4 * c + 2 * half;
            v2f a, b;
            a.x = at[m * 129 + k0];
            a.y = at[m * 129 + k0 + 1];
            b.x = w0[(size_t)k0 * HF + ncol];
            b.y = w0[(size_t)(k0 + 1) * HF + ncol];
            acc = __builtin_amdgcn_wmma_f32_16x16x4_f32(
                false, a, false, b, (short)0, acc, false, false);
        }
        float bias = b0[ncol];
        for (int r = 0; r < 8; ++r) {
            int mrow = rowt * 16 + r + 8 * half;
            h0[(size_t)(node0 + mrow) * HF + ncol] = acc[r] + bias;
        }
    }
}

// ---------------- Fused SGConv layer ----------------
// block: 256 threads (8 waves), 32 nodes. Phase 1: cooperative gather-agg.
// Phase 2: waves 0-3 do the 32x32 GEMM via 8 chained f32 WMMAs + bias + ReLU.

__global__ void layer_kernel(const float* __restrict__ h,
                             const int* __restrict__ row_ptr,
                             const int2* __restrict__ edge,
                             const float* __restrict__ dinv,
                             const float* __restrict__ W,
                             const float* __restrict__ bias,
                             float* __restrict__ h_out) {
    __shared__ float tile[32 * 33];    // 32x32 agg tile, padded stride 33
    int t = threadIdx.x;
    int wave = t >> 5, lane = t & 31;
    int node0 = blockIdx.x * 32;

    // Phase 1: each wave aggregates 4 nodes; lane == feature (H == wave32).
    for (int j = 0; j < 4; ++j) {
        int nl = (wave << 2) + j;
        int i  = node0 + nl;
        float di  = dinv[i];
        float acc = h[(size_t)i * HF + lane] * di * di;   // self loop
        int beg = row_ptr[i], end = row_ptr[i + 1];

        int kb = beg;
        // Full 32-edge chunks: one coalesced b64 load per lane, then
        // broadcast each edge's (src, w) with v_readlane -> SGPR, so the
        // gather becomes scalar-base + constant lane offset: 1 vmem/edge.
        for (; kb + 32 <= end; kb += 32) {
            int2 ew = edge[kb + lane];
            if (kb + 32 < end) __builtin_prefetch(&edge[kb + 32 + lane], 0, 1);
            #pragma unroll 8
            for (int u = 0; u < 32; ++u) {
                int   sj = __builtin_amdgcn_readlane(ew.x, u);
                float wj = __int_as_float(__builtin_amdgcn_readlane(ew.y, u));
                const float* hp = h + (size_t)sj * HF;     // scalar base
                acc = fmaf(hp[lane], wj, acc);
            }
        }
        // Tail (< 32 edges).
        if (kb < end) {
            int kk = kb + lane;
            int2 ew = (kk < end) ? edge[kk] : make_int2(0, 0);
            int cnt = end - kb;
            for (int u = 0; u < cnt; ++u) {
                int   sj = __builtin_amdgcn_readlane(ew.x, u);
                float wj = __int_as_float(__builtin_amdgcn_readlane(ew.y, u));
                const float* hp = h + (size_t)sj * HF;
                acc = fmaf(hp[lane], wj, acc);
            }
        }
        tile[nl * 33 + lane] = acc;
    }
    __syncthreads();

    // Phase 2: (32x32 agg) @ (32x32 W) -> four 16x16 output tiles.
    if (wave < 4) {
        int rowt = wave >> 1;
        int colh = wave & 1;
        int half = lane >> 4;
        int m    = lane & 15;
        int ncol = (colh << 4) + m;
        const float* at = tile + rowt * 16 * 33;
        v8f acc = {};
        for (int c = 0; c < HF / 4; ++c) {
            int k0 = 4 * c + 2 * half;
            v2f a, b;
            a.x = at[m * 33 + k0];
            a.y = at[m * 33 + k0 + 1];
            b.x = W[k0 * HF + ncol];
            b.y = W[(k0 + 1) * HF + ncol];
            acc = __builtin_amdgcn_wmma_f32_16x16x4_f32(
                false, a, false, b, (short)0, acc, false, false);
        }
        float bb = bias[ncol];
        for (int r = 0; r < 8; ++r) {
            int mrow = rowt * 16 + r + 8 * half;
            float v = acc[r] + bb;
            h_out[(size_t)(node0 + mrow) * HF + ncol] = v > 0.f ? v : 0.f;
        }
    }
}

// ---------------- Output GEMM: out = h @ w16 + b16 ----------------
// block: 256 threads, 32 nodes, 8 waves -> eight 16x16 output tiles (64 cols).

__global__ void gemm_out_kernel(const float* __restrict__ h,
                                const float* __restrict__ w16,
                                const float* __restrict__ b16,
                                float* __restrict__ out) {
    __shared__ float tile[32 * 33];
    int t = threadIdx.x;
    int node0 = blockIdx.x * 32;
    for (int idx = t; idx < 32 * HF; idx += 256) {
        int r = idx >> 5, c = idx & 31;
        tile[r * 33 + c] = h[(size_t)(node0 + r) * HF + c];
    }
    __syncthreads();
    int wave = t >> 5, lane = t & 31;
    int rowt = wave >> 2;                  // 0..1
    int colt = wave & 3;                   // 0..3
    int half = lane >> 4;
    int m    = lane & 15;
    int ncol = (colt << 4) + m;            // 0..63
    const float* at = tile + rowt * 16 * 33;
    v8f acc = {};
    for (int c = 0; c < HF / 4; ++c) {
        int k0 = 4 * c + 2 * half;
        v2f a, b;
        a.x = at[m * 33 + k0];
        a.y = at[m * 33 + k0 + 1];
        b.x = w16[(size_t)k0 * DOUT + ncol];
        b.y = w16[(size_t)(k0 + 1) * DOUT + ncol];
        acc = __builtin_amdgcn_wmma_f32_16x16x4_f32(
            false, a, false, b, (short)0, acc, false, false);
    }
    float bb = b16[ncol];
    for (int r = 0; r < 8; ++r) {
        int mrow = rowt * 16 + r + 8 * half;
        out[(size_t)(node0 + mrow) * DOUT + ncol] = acc[r] + bb;
    }
}

// ---------------- Host launch ----------------

static inline size_t align256(size_t x) { return (x + 255) & ~(size_t)255; }

extern "C" void kernel_launch(void* const* d_in, const int* in_sizes, int n_in,
                              void* d_out, int out_size, void* d_ws, size_t ws_size,
                              hipStream_t stream) {
    (void)n_in; (void)out_size; (void)ws_size;
    const float* x   = (const float*)d_in[0];
    const int*   ei  = (const int*)d_in[1];      // (2, E): src row then dst row
    const float* w0  = (const float*)d_in[2];
    const float* b0  = (const float*)d_in[3];
    const float* Wl  = (const float*)d_in[4];    // (L, 32, 32)
    const float* bl  = (const float*)d_in[5];    // (L, 32)
    const float* w16 = (const float*)d_in[6];
    const float* b16 = (const float*)d_in[7];

    const int N = in_sizes[0] / DIN;             // 100000 (multiple of 32)
    const int E = in_sizes[1] / 2;               // 3200000

    // Workspace carve-out.
    char* ws = (char*)d_ws;
    size_t off = 0;
    int*   cnt      = (int*)(ws + off);   off = align256(off + (size_t)N * 4);
    float* dinv     = (float*)(ws + off); off = align256(off + (size_t)N * 4);
    int*   row_ptr  = (int*)(ws + off);   off = align256(off + (size_t)(N + 1) * 4);
    int*   fill_ptr = (int*)(ws + off);   off = align256(off + (size_t)N * 4);
    int2*  edge     = (int2*)(ws + off);  off = align256(off + (size_t)E * 8);
    float* hA       = (float*)(ws + off); off = align256(off + (size_t)N * HF * 4);
    float* hB       = (float*)(ws + off); off = align256(off + (size_t)N * HF * 4);

    const int nb_n = (N + 255) / 256;
    const int nb_e = (E + 255) / 256;
    const int nb_t = N / 32;                     // 32-node tiles

    // CSR build (re-done each call; resets all mutable workspace state).
    zero_cnt_kernel<<<nb_n, 256, 0, stream>>>(cnt, N);
    count_kernel<<<nb_e, 256, 0, stream>>>(ei, cnt, E);
    dinv_kernel<<<nb_n, 256, 0, stream>>>(cnt, dinv, N);
    scan_kernel<<<1, 256, 0, stream>>>(cnt, row_ptr, fill_ptr, N);
    fill_kernel<<<nb_e, 256, 0, stream>>>(ei, dinv, fill_ptr, edge, E);

    // h0 = x @ w0 + b0
    gemm_in_kernel<<<nb_t, 256, 0, stream>>>(x, w0, b0, hA);

    // 16 fused SGConv layers (ping-pong h buffers).
    float* hin = hA;
    float* hout = hB;
    for (int l = 0; l < NLAYERS; ++l) {
        layer_kernel<<<nb_t, 256, 0, stream>>>(hin, row_ptr, edge, dinv,
                                               Wl + (size_t)l * HF * HF,
                                               bl + (size_t)l * HF, hout);
        float* tmp = hin; hin = hout; hout = tmp;
    }

    // out = h @ w16 + b16
    gemm_out_kernel<<<nb_t, 256, 0, stream>>>(hin, w16, b16, (float*)d_out);
}